// EmbProb_67748814127432
// MI455X (gfx1250) — compile-verified
//
#include <hip/hip_runtime.h>

// ---------------------------------------------------------------------------
// EmbProb mask-fill: out[i, j] = (j >= k_i) ? 1.0f : 0.0f
//   k_i = (int)(round(x_i * 8) / 8 * 4096 * 0.5) = rint(x_i * 8) * 256
// Output: 8192 x 4096 fp32 = 128 MiB  -> pure store-bandwidth kernel
// (23.3 TB/s => ~5.8 us floor). No contraction => WMMA inapplicable; the
// CDNA5 feature that fits is the async LDS->global store engine (ASYNCcnt).
//
// Since k is always a multiple of 256, every row is a shifted 4096-wide
// window of one 6144-float prototype P[t] = (t >= 2048). We build P in LDS
// once per block, then stream each row out with GLOBAL_STORE_ASYNC_FROM_LDS
// at LDS offset (2048 - k_i)*4 bytes (always 1 KiB aligned).
//
// Toolchain probe result (round 1 diagnostic): the async builtin's params are
// typed 128-bit int-vector pointers, AS1 (printed "__device__") for global
// and AS3 for LDS — not void*.
// ---------------------------------------------------------------------------

#ifndef __has_builtin
#define __has_builtin(x) 0
#endif

#if __has_builtin(__builtin_amdgcn_global_store_async_from_lds_b128)
#define EMB_ASYNC 1
#else
#define EMB_ASYNC 0
#endif

#define EMB_C     4096
#define EMB_HALF  2048
#define EMB_PROTO (EMB_C + EMB_HALF)   // 6144 floats = 24 KiB LDS
#define EMB_ROWS  8192
#define EMB_TPB   256
#define EMB_RPB   8                    // rows per block

typedef float f4  __attribute__((ext_vector_type(4)));
typedef int   v4i __attribute__((ext_vector_type(4)));
// Pointer-to-AS1 v4i and pointer-to-AS3 v4i (pointers themselves generic).
typedef __attribute__((address_space(1))) v4i* g_v4i_p;
typedef __attribute__((address_space(3))) v4i* l_v4i_p;

__global__ __launch_bounds__(EMB_TPB)
void embprob_mask_kernel(const float* __restrict__ x,
                         float* __restrict__ out,
                         int n_rows, int rows_per_block) {
#if EMB_ASYNC
    __shared__ float proto[EMB_PROTO];
    // Prototype step function: 2048 zeros then 4096 ones.
    for (int t = (int)threadIdx.x; t < EMB_PROTO; t += EMB_TPB)
        proto[t] = (t >= EMB_HALF) ? 1.0f : 0.0f;
    __syncthreads();   // LDS writes visible before async engine reads them
#endif

    const int row0 = (int)blockIdx.x * rows_per_block;
    for (int r = 0; r < rows_per_block; ++r) {
        const int i = row0 + r;
        if (i >= n_rows) break;

        // Uniform per-row threshold. rintf = round-to-nearest-even, matching
        // jnp.round; round(x*8)/8 * 2048 == round(x*8) * 256 exactly.
        const float xi = x[i];
        const int   k  = ((int)rintf(xi * 8.0f)) << 8;   // 0..2048, step 256

        float* dst = out + (size_t)i * EMB_C;

#if EMB_ASYNC
        const int shift = EMB_HALF - k;                  // 0..2048, step 256
        // 4096 floats / 256 threads = 4 x b128 async transfers per thread.
        #pragma unroll
        for (int it = 0; it < 4; ++it) {
            const int col = (int)threadIdx.x * 4 + it * 1024;  // 16B aligned
            __builtin_amdgcn_global_store_async_from_lds_b128(
                (g_v4i_p)(dst + col),
                (l_v4i_p)(proto + shift + col),
                /*offset=*/0, /*cpol=*/0);
        }
#else
        // Fallback: direct coalesced b128 stores of the constant pattern.
        // k is a multiple of 256, so each float4 is uniform.
        #pragma unroll
        for (int it = 0; it < 4; ++it) {
            const int col = (int)threadIdx.x * 4 + it * 1024;
            const float v = (col >= k) ? 1.0f : 0.0f;
            *(f4*)(dst + col) = (f4){v, v, v, v};
        }
#endif
    }

#if EMB_ASYNC && __has_builtin(__builtin_amdgcn_s_wait_asynccnt)
    // Drain outstanding async LDS->global stores before wave exit
    // (S_ENDPGM also does an implicit wait-idle, this makes it explicit).
    __builtin_amdgcn_s_wait_asynccnt(0);
#endif
}

extern "C" void kernel_launch(void* const* d_in, const int* in_sizes, int n_in,
                              void* d_out, int out_size, void* d_ws, size_t ws_size,
                              hipStream_t stream) {
    const float* x   = (const float*)d_in[0];
    float*       out = (float*)d_out;

    const int n_rows = (n_in > 0 && in_sizes) ? in_sizes[0] : EMB_ROWS;  // 8192
    const int rpb    = EMB_RPB;
    const int blocks = (n_rows + rpb - 1) / rpb;                          // 1024

    embprob_mask_kernel<<<blocks, EMB_TPB, 0, stream>>>(x, out, n_rows, rpb);
}